// DistanceWeightedDiceLoss_7576322310176
// MI455X (gfx1250) — compile-verified
//
#include <hip/hip_runtime.h>

// Problem geometry: (N=4, C=1, Z=160, Y=160, X=160) float32, row-major, x innermost.
#define DX     160
#define SY     160          // y stride (elements)
#define SZ     25600        // z stride
#define SN     4096000      // batch stride
#define NVOX   16384000LL   // 4*160^3
#define RAD    9            // 9 iterations of 3-min-pool == radius-9 min filter
#define SIGMA  5.0f
#define SMOOTH 1e-5f
#define NBLK2  3200         // 4 * 160 * 5 tiles for K2/K3

// ---------------- CDNA5 feature probes ----------------
#if __has_builtin(__builtin_amdgcn_global_load_async_to_lds_b128)
#define HAS_ASYNC_LDS 1
#else
#define HAS_ASYNC_LDS 0
#endif
#if __has_builtin(__builtin_amdgcn_wmma_f32_16x16x4_f32)
#define HAS_WMMA_F32X4 1
#else
#define HAS_WMMA_F32X4 0
#endif

typedef __attribute__((ext_vector_type(4))) int   v4i;
typedef __attribute__((ext_vector_type(2))) float v2f;
typedef __attribute__((ext_vector_type(8))) float v8f;

// Exact parameter types of the async builtin (from the round-1 diagnostic):
//   (v4i addrspace(1)*, v4i addrspace(3)*, imm int offset, imm int cpol)
typedef __attribute__((address_space(1))) v4i* gv4p;   // pointer to AS1 v4i
typedef __attribute__((address_space(3))) v4i* lv4p;   // pointer to AS3 v4i

// 16B global -> LDS copy; async CDNA5 path (ASYNCcnt) with synchronous fallback.
__device__ __forceinline__ void cp16_g2l(const float* g, float* l) {
#if HAS_ASYNC_LDS
  __builtin_amdgcn_global_load_async_to_lds_b128(
      (gv4p)(unsigned long long)g,   // flat global addr bit-pattern == AS1 addr
      (lv4p)l,                       // generic -> LDS addrspace cast (keeps LDS offset)
      0, 0);
#else
  *(float4*)l = *(const float4*)g;
#endif
}

__device__ __forceinline__ void cp_wait_barrier() {
#if HAS_ASYNC_LDS
#if __has_builtin(__builtin_amdgcn_s_wait_asynccnt)
  __builtin_amdgcn_s_wait_asynccnt(0);
#else
  asm volatile("s_wait_asynccnt 0" ::: "memory");
#endif
#endif
  __syncthreads();
}

// Wave32 sum via V_WMMA_F32_16X16X4_F32.
// A: only row 0 nonzero (ones)  => D[0][n] = B[0][n] + B[2][n] = v[n] + v[n+16]
// (B 32-bit layout: lane<16 -> B[0][lane]=b.x ; lane>=16 -> B[2][lane-16]=b.x)
// D vgpr0: lanes 0..15 hold D[0][n]; finish with 4 xor-shuffles. Lane 0 holds the sum.
__device__ __forceinline__ float wave_sum32(float v) {
#if HAS_WMMA_F32X4
  const int lane = threadIdx.x & 31;
  const float sel = ((lane & 15) == 0) ? 1.0f : 0.0f;
  v2f a; a[0] = sel; a[1] = sel;
  v2f b; b[0] = v;   b[1] = 0.0f;
  v8f c = {};
  v8f d = __builtin_amdgcn_wmma_f32_16x16x4_f32(false, a, false, b, (short)0, c, false, false);
  float s = d[0];
  s += __shfl_xor(s, 1, 32);
  s += __shfl_xor(s, 2, 32);
  s += __shfl_xor(s, 4, 32);
  s += __shfl_xor(s, 8, 32);
  return s;                       // valid in lane 0 (and lanes 1..15)
#else
  for (int m = 16; m; m >>= 1) v += __shfl_xor(v, m, 32);
  return v;
#endif
}

// -------- K1: radius-9 min along x, target -> ws (one 160-row per block) --------
__global__ void __launch_bounds__(160) k_minx(const float* __restrict__ in,
                                              float* __restrict__ out) {
  __shared__ float row[DX];
  const long long base = (long long)blockIdx.x * DX;
  const int i = threadIdx.x;
  row[i] = in[base + i];
  __syncthreads();
  int lo = i - RAD; if (lo < 0) lo = 0;
  int hi = i + RAD; if (hi > DX - 1) hi = DX - 1;
  float m = row[lo];
  for (int j = lo + 1; j <= hi; ++j) m = fminf(m, row[j]);
  out[base + i] = m;
}

// -------- K2: radius-9 min along y, in-place on ws --------
// Tile = 32 x-columns (one 128B line) x full y extent (160) = 20KB LDS snapshot.
__global__ void __launch_bounds__(256) k_miny(float* __restrict__ buf) {
  __shared__ __align__(16) float tile[DX * 32];
  const long long base = (long long)blockIdx.z * SN + (long long)blockIdx.y * SZ
                       + blockIdx.x * 32;
#pragma unroll
  for (int k = 0; k < 5; ++k) {                       // 1280 vec4s / 256 threads
    const int v = threadIdx.x + k * 256;
    const int r = v >> 3, c4 = v & 7;                 // r = y, c4 = vec4-in-row
    cp16_g2l(&buf[base + (long long)r * SY + c4 * 4], &tile[v * 4]);
  }
  cp_wait_barrier();
  for (int o = threadIdx.x; o < DX * 32; o += 256) {  // 20 outputs/thread
    const int r = o >> 5, c = o & 31;
    int lo = r - RAD; if (lo < 0) lo = 0;
    int hi = r + RAD; if (hi > DX - 1) hi = DX - 1;
    float m = tile[lo * 32 + c];
    for (int j = lo + 1; j <= hi; ++j) m = fminf(m, tile[j * 32 + c]);
    buf[base + (long long)r * SY + c] = m;
  }
}

// -------- K3: radius-9 min along z + weights + block reduction --------
__global__ void __launch_bounds__(256) k_minz_reduce(
    const float* __restrict__ ebuf, const float* __restrict__ pred,
    const float* __restrict__ target, float* __restrict__ pI,
    float* __restrict__ pP, float* __restrict__ pT) {
  __shared__ __align__(16) float tile[DX * 32];
  const long long base = (long long)blockIdx.z * SN + (long long)blockIdx.y * SY
                       + blockIdx.x * 32;
#pragma unroll
  for (int k = 0; k < 5; ++k) {
    const int v = threadIdx.x + k * 256;
    const int r = v >> 3, c4 = v & 7;                 // r = z
    cp16_g2l(&ebuf[base + (long long)r * SZ + c4 * 4], &tile[v * 4]);
  }
  cp_wait_barrier();
  float aI = 0.f, aP = 0.f, aT = 0.f;
  for (int o = threadIdx.x; o < DX * 32; o += 256) {  // exactly 20 per thread, uniform
    const int r = o >> 5, c = o & 31;
    int lo = r - RAD; if (lo < 0) lo = 0;
    int hi = r + RAD; if (hi > DX - 1) hi = DX - 1;
    float e = tile[lo * 32 + c];
    for (int j = lo + 1; j <= hi; ++j) e = fminf(e, tile[j * 32 + c]);
    const long long g = base + (long long)r * SZ + c;
    const float t = target[g];
    const float p = __builtin_nontemporal_load(&pred[g]);   // single-use stream
    const float w = fmaf(t - e, SIGMA, 1.0f);
    aI = fmaf(w * p, t, aI);
    aP = fmaf(w, p, aP);
    aT = fmaf(w, t, aT);
  }
  // EXEC is all-ones here (uniform loop), safe for WMMA-based wave reduction.
  aI = wave_sum32(aI);
  aP = wave_sum32(aP);
  aT = wave_sum32(aT);
  __shared__ float rI[8], rP[8], rT[8];
  const int lane = threadIdx.x & 31, wv = threadIdx.x >> 5;
  if (lane == 0) { rI[wv] = aI; rP[wv] = aP; rT[wv] = aT; }
  __syncthreads();
  if (threadIdx.x == 0) {
    float sI = 0.f, sP = 0.f, sT = 0.f;
    for (int q = 0; q < 8; ++q) { sI += rI[q]; sP += rP[q]; sT += rT[q]; }
    const int bid = ((int)blockIdx.z * (int)gridDim.y + (int)blockIdx.y) * (int)gridDim.x
                  + (int)blockIdx.x;
    pI[bid] = sI; pP[bid] = sP; pT[bid] = sT;
  }
}

// -------- K4: 3200 partials -> scalar loss --------
__global__ void __launch_bounds__(256) k_final(const float* __restrict__ pI,
                                               const float* __restrict__ pP,
                                               const float* __restrict__ pT,
                                               int nblk, float* __restrict__ out) {
  float aI = 0.f, aP = 0.f, aT = 0.f;
  for (int i = threadIdx.x; i < nblk; i += 256) { aI += pI[i]; aP += pP[i]; aT += pT[i]; }
  aI = wave_sum32(aI);
  aP = wave_sum32(aP);
  aT = wave_sum32(aT);
  __shared__ float rI[8], rP[8], rT[8];
  const int lane = threadIdx.x & 31, wv = threadIdx.x >> 5;
  if (lane == 0) { rI[wv] = aI; rP[wv] = aP; rT[wv] = aT; }
  __syncthreads();
  if (threadIdx.x == 0) {
    float I = 0.f, P = 0.f, T = 0.f;
    for (int q = 0; q < 8; ++q) { I += rI[q]; P += rP[q]; T += rT[q]; }
    const float dice = (2.0f * I + SMOOTH) / (P + T + SMOOTH);
    out[0] = 1.0f - dice;
  }
}

extern "C" void kernel_launch(void* const* d_in, const int* in_sizes, int n_in,
                              void* d_out, int out_size, void* d_ws, size_t ws_size,
                              hipStream_t stream) {
  (void)in_sizes; (void)n_in; (void)out_size; (void)ws_size;
  const float* pred   = (const float*)d_in[0];
  const float* target = (const float*)d_in[1];
  float* out = (float*)d_out;

  float* ws1 = (float*)d_ws;          // eroded buffer: 65.5 MB (L2-resident on MI455X)
  float* pI  = ws1 + NVOX;            // 3*3200 f32 partials
  float* pP  = pI + NBLK2;
  float* pT  = pP + NBLK2;

  // x-filter: one block per contiguous 160-row (4*160*160 rows)
  k_minx<<<dim3(102400), dim3(160), 0, stream>>>(target, ws1);
  // y-filter, in-place: tiles over (xtile=5, z=160, n=4)
  k_miny<<<dim3(5, 160, 4), dim3(256), 0, stream>>>(ws1);
  // z-filter + fused weighted-dice partial reduction: tiles over (xtile=5, y=160, n=4)
  k_minz_reduce<<<dim3(5, 160, 4), dim3(256), 0, stream>>>(ws1, pred, target, pI, pP, pT);
  // final scalar
  k_final<<<1, 256, 0, stream>>>(pI, pP, pT, NBLK2, out);
}